// VarQEC_73289321938948
// MI455X (gfx1250) — compile-verified
//
#include <hip/hip_runtime.h>
#include <math.h>

#define N_QUBIT 22
#define N_GATES 44
#define DIM     (1u << N_QUBIT)      // 4,194,304 amplitudes per logical row
#define NPAIRS  (DIM / 2u)           // 2^21 amplitude pairs per qubit

typedef __attribute__((ext_vector_type(2))) float v2f;
typedef __attribute__((ext_vector_type(8))) float v8f;

// ---------------------------------------------------------------------------
// Init: psi[0][0] = 1, psi[1][1] = 1, everything else 0; zero Gram accumulators
// ---------------------------------------------------------------------------
__global__ void vq_init(float2* __restrict__ psi, float* __restrict__ gAcc) {
    unsigned t = blockIdx.x * blockDim.x + threadIdx.x;
    if (t < 2u * DIM) {
        float2 v = {0.f, 0.f};
        if (t == 0u)        v.x = 1.f;   // row0, state |0...0>
        if (t == DIM + 1u)  v.x = 1.f;   // row1, state |0...01>
        psi[t] = v;
    }
    if (t < (unsigned)(N_QUBIT * 64)) gAcc[t] = 0.f;
}

// ---------------------------------------------------------------------------
// Precompute (cos(theta/2), sin(theta/2)) for all 44 RY gates
// ---------------------------------------------------------------------------
__global__ void vq_trig(const float* __restrict__ theta, float2* __restrict__ cs) {
    int t = threadIdx.x;
    if (t < N_GATES) {
        float h = 0.5f * theta[t];
        cs[t] = make_float2(cosf(h), sinf(h));
    }
}

// ---------------------------------------------------------------------------
// RY(theta) on qubit with pair-stride m = 2^(21-q).
// Streaming pass: each thread owns one (s0, s0+m) pair of one logical row.
// ---------------------------------------------------------------------------
__global__ void vq_ry(float2* __restrict__ psi, const float2* __restrict__ cs,
                      int g, unsigned m) {
    unsigned t   = blockIdx.x * blockDim.x + threadIdx.x;   // [0, DIM)
    unsigned row = t >> (N_QUBIT - 1);
    unsigned p   = t & (NPAIRS - 1u);
    unsigned s0  = ((p & ~(m - 1u)) << 1) | (p & (m - 1u));
    float2 cse = cs[g];
    float c = cse.x, s = cse.y;
    float2* b = psi + (size_t)row * DIM;
    float2 a0 = b[s0];
    float2 a1 = b[s0 + m];
    float2 o0 = { c * a0.x - s * a1.x, c * a0.y - s * a1.y };
    float2 o1 = { s * a0.x + c * a1.x, s * a0.y + c * a1.y };
    b[s0]     = o0;
    b[s0 + m] = o1;
}

// ---------------------------------------------------------------------------
// CNOT: control bit-position pc, target bit-position pt (bit k has stride 2^k).
// For every index with bit pc = 1 and bit pt = 0, swap with partner (bit pt = 1).
// One thread per swap pair: 2 rows * DIM/4 threads.
// ---------------------------------------------------------------------------
__global__ void vq_cnot(float2* __restrict__ psi, int pc, int pt) {
    unsigned t   = blockIdx.x * blockDim.x + threadIdx.x;   // [0, 2*DIM/4)
    unsigned row = t >> (N_QUBIT - 2);
    unsigned p   = t & ((DIM / 4u) - 1u);
    int lo = pc < pt ? pc : pt;
    int hi = pc < pt ? pt : pc;
    unsigned low  = p & ((1u << lo) - 1u);
    unsigned mid  = (p >> lo) & ((1u << (hi - 1 - lo)) - 1u);
    unsigned high = p >> (hi - 1);
    unsigned s = (high << (hi + 1)) | (mid << (lo + 1)) | low;  // bits lo,hi clear
    s |= (1u << pc);                                            // control = 1
    unsigned s1 = s | (1u << pt);                               // target partner
    float2* b = psi + (size_t)row * DIM;
    float2 tmp = b[s];
    b[s]  = b[s1];
    b[s1] = tmp;
}

// ---------------------------------------------------------------------------
// Fused X/Y/Z measurement for one qubit via an 8x8 Gram matrix accumulated
// with V_WMMA_F32_16X16X4_F32.
//
// For each pair (s0, s1=s0+mq), define the 8-vector
//   v = (Re psi0[s0], Im psi0[s0], Re psi1[s0], Im psi1[s0],
//        Re psi0[s1], Im psi0[s1], Re psi1[s1], Im psi1[s1]).
// G = sum_pairs v v^T contains every real dot product needed for
// T_X, T_Y, T_Z of this qubit.
//
// WMMA 16x16x4 layouts (wave32):
//   A 16x4 : lanes 0-15 hold M=lane, K={0,1}; lanes 16-31 hold K={2,3}.
//   B 4x16 : lanes 0-15 hold N=lane, K={0,1}; lanes 16-31 hold K={2,3}.
// With A[m][k] = v_m[pair k] and B[k][n] = v_n[pair k], the per-lane register
// contents of A and B are identical -> a single v2f feeds both operands.
// K accumulates across the grid-stride loop in C (D = A*B + C).
// Rows/cols 8..15 are zero. Loop trip count is wave-uniform -> EXEC all-1s
// at every WMMA as the ISA requires.
// ---------------------------------------------------------------------------
__device__ __forceinline__ unsigned vq_pair_addr(unsigned p, unsigned mq, bool useS1) {
    unsigned s0 = ((p & ~(mq - 1u)) << 1) | (p & (mq - 1u));
    return useS1 ? (s0 + mq) : s0;
}

__global__ void vq_reduce(const float* __restrict__ psiF,   // psi as raw floats
                          float* __restrict__ gAccQ,        // 64 floats for this qubit
                          unsigned mq) {
    unsigned lane = threadIdx.x & 31u;
    unsigned wid  = (blockIdx.x * blockDim.x + threadIdx.x) >> 5;
    unsigned nw   = (gridDim.x * blockDim.x) >> 5;

    unsigned mcomp = lane & 15u;         // which of the 16 A-rows / B-cols
    unsigned kgrp  = lane >> 4;          // 0 -> pairs {0,1}, 1 -> pairs {2,3}
    bool     valid = mcomp < 8u;         // only 8 components are real data
    unsigned i     = (mcomp >> 1) & 1u;  // logical row 0/1
    bool     useS1 = mcomp >= 4u;        // s0 half vs s1 half
    unsigned isIm  = mcomp & 1u;         // Re / Im

    const float* basef = psiF + (size_t)i * DIM * 2u + isIm;

    v8f acc = {0.f, 0.f, 0.f, 0.f, 0.f, 0.f, 0.f, 0.f};

    for (unsigned pb = wid * 4u; pb < NPAIRS; pb += nw * 4u) {
        float a0 = 0.f, a1 = 0.f;
        if (valid) {
            unsigned p0 = pb + kgrp * 2u;
            a0 = basef[(size_t)vq_pair_addr(p0,      mq, useS1) * 2u];
            a1 = basef[(size_t)vq_pair_addr(p0 + 1u, mq, useS1) * 2u];
        }
        v2f av = {a0, a1};
        acc = __builtin_amdgcn_wmma_f32_16x16x4_f32(
                  false, av, false, av, (short)0, acc, false, false);
    }

    // D layout: VGPR r, lanes 0-15 -> G[r][lane]. We need G[0..7][0..7].
    if (lane < 8u) {
#pragma unroll
        for (int r = 0; r < 8; ++r)
            atomicAdd(&gAccQ[r * 8 + (int)lane], acc[r]);
    }
}

// ---------------------------------------------------------------------------
// Finalize: per qubit build T_X, T_Y, T_Z (2x2 complex) from the Gram matrix,
// subtract lam*I (lam = mean of diagonal), accumulate |res|^2 over all 66 ops.
// One wave32; lane q handles qubit q; shuffle-reduce; lane 0 writes the loss.
// ---------------------------------------------------------------------------
__device__ __forceinline__ float vq_op_loss(const float Tr[2][2], const float Ti[2][2]) {
    float lr = 0.5f * (Tr[0][0] + Tr[1][1]);
    float li = 0.5f * (Ti[0][0] + Ti[1][1]);
    float d0r = Tr[0][0] - lr, d0i = Ti[0][0] - li;
    float d1r = Tr[1][1] - lr, d1i = Ti[1][1] - li;
    return d0r * d0r + d0i * d0i + d1r * d1r + d1i * d1i
         + Tr[0][1] * Tr[0][1] + Ti[0][1] * Ti[0][1]
         + Tr[1][0] * Tr[1][0] + Ti[1][0] * Ti[1][0];
}

__global__ void vq_final(const float* __restrict__ gAcc, float* __restrict__ out) {
    int lane = threadIdx.x;
    float loss = 0.f;
    if (lane < N_QUBIT) {
        const float* G = gAcc + lane * 64;
        // U[i][j] = sum conj(psi_i[s0]) * psi_j[s1]
        float Ur[2][2], Ui[2][2];
#pragma unroll
        for (int i = 0; i < 2; ++i)
#pragma unroll
            for (int j = 0; j < 2; ++j) {
                Ur[i][j] = G[(2 * i) * 8 + 4 + 2 * j] + G[(2 * i + 1) * 8 + 5 + 2 * j];
                Ui[i][j] = G[(2 * i) * 8 + 5 + 2 * j] - G[(2 * i + 1) * 8 + 4 + 2 * j];
            }
        float TXr[2][2], TXi[2][2], TYr[2][2], TYi[2][2], TZr[2][2], TZi[2][2];
#pragma unroll
        for (int i = 0; i < 2; ++i)
#pragma unroll
            for (int j = 0; j < 2; ++j) {
                // T_X = U + conj(U^T)
                TXr[i][j] = Ur[i][j] + Ur[j][i];
                TXi[i][j] = Ui[i][j] - Ui[j][i];
                // T_Y = i * (conj(U^T) - U)
                TYr[i][j] = Ui[j][i] + Ui[i][j];
                TYi[i][j] = Ur[j][i] - Ur[i][j];
                // T_Z = <x_i|x_j> - <y_i|y_j>
                TZr[i][j] = (G[(2 * i) * 8 + 2 * j]     + G[(2 * i + 1) * 8 + 2 * j + 1])
                          - (G[(4 + 2 * i) * 8 + 4 + 2 * j] + G[(5 + 2 * i) * 8 + 5 + 2 * j]);
                TZi[i][j] = (G[(2 * i) * 8 + 2 * j + 1] - G[(2 * i + 1) * 8 + 2 * j])
                          - (G[(4 + 2 * i) * 8 + 5 + 2 * j] - G[(5 + 2 * i) * 8 + 4 + 2 * j]);
            }
        loss = vq_op_loss(TXr, TXi) + vq_op_loss(TYr, TYi) + vq_op_loss(TZr, TZi);
    }
#pragma unroll
    for (int k = 16; k > 0; k >>= 1)
        loss += __shfl_down(loss, k, 32);
    if (lane == 0) out[0] = loss;
}

// ---------------------------------------------------------------------------
// Host-side orchestration (graph-capture safe: fixed launches, no allocation)
// ---------------------------------------------------------------------------
extern "C" void kernel_launch(void* const* d_in, const int* in_sizes, int n_in,
                              void* d_out, int out_size, void* d_ws, size_t ws_size,
                              hipStream_t stream) {
    const float* theta = (const float*)d_in[0];   // [2][22] float32
    float* out = (float*)d_out;                   // scalar float32 loss

    char*  base      = (char*)d_ws;
    size_t psi_bytes = (size_t)2 * DIM * sizeof(float2);   // 64 MB state
    float2* psi  = (float2*)base;
    float2* cs   = (float2*)(base + psi_bytes);            // 44 (cos,sin) pairs
    float*  gAcc = (float*)(base + psi_bytes + 1024);      // 22 * 64 Gram floats

    vq_init<<<(2u * DIM + 255u) / 256u, 256, 0, stream>>>(psi, gAcc);
    vq_trig<<<1, 64, 0, stream>>>(theta, cs);

    for (int l = 0; l < 2; ++l) {
        for (int q = 0; q < N_QUBIT; ++q)
            vq_ry<<<DIM / 256u, 256, 0, stream>>>(
                psi, cs, l * N_QUBIT + q, 1u << (N_QUBIT - 1 - q));
        for (int q = 0; q < N_QUBIT; ++q) {
            int c = q, t = (q + 1) % N_QUBIT;
            vq_cnot<<<(2u * (DIM / 4u)) / 256u, 256, 0, stream>>>(
                psi, N_QUBIT - 1 - c, N_QUBIT - 1 - t);
        }
    }

    for (int q = 0; q < N_QUBIT; ++q)
        vq_reduce<<<512, 256, 0, stream>>>(
            (const float*)psi, gAcc + q * 64, 1u << (N_QUBIT - 1 - q));

    vq_final<<<1, 32, 0, stream>>>(gAcc, out);
}